// DLRMNet_76003741270612
// MI455X (gfx1250) — compile-verified
//
#include <hip/hip_runtime.h>

// ---------------------------------------------------------------------------
// DLRM on MI455X (gfx1250, wave32, WMMA).
// Dominant cost: fused interaction + top-MLP layer-0 GEMM,
// M=1024 N=512 K=93544 fp32 (~98 GFLOP), memory-bound on w_t0 (191.6 MB ~= L2).
// The 383 MB top_in matrix is never materialized: the interaction A-operand is
// generated on the fly from a 64-row feature stage in LDS. fp32 WMMA 16x16x4.
// Blocking: 4 M-tiles per wave (register reuse of B) x 4 N-tiles per block
// (LDS reuse of features); w stream prefetched ~2KB ahead into L0/L2.
// ---------------------------------------------------------------------------

typedef float v2f __attribute__((ext_vector_type(2)));
typedef float v8f __attribute__((ext_vector_type(8)));

#define BATCH   1024
#define NDENSE  13
#define NTAB    26
#define VOCAB   100000
#define DEMB    16
#define FDIM    432                 // 16 + 26*16
#define NPAIR   93528               // FDIM*(FDIM+1)/2
#define KTOP0   93544               // 16 + NPAIR
#define MT      4                   // M-tiles per wave in fused kernel

// ---------------------------------------------------------------------------
// Pair index table: p -> (i,j) for row-major upper-triangle (i<=j) of FDIM.
// ---------------------------------------------------------------------------
__global__ __launch_bounds__(256)
void init_pairs_kernel(unsigned int* __restrict__ pairs) {
    int p = blockIdx.x * blockDim.x + threadIdx.x;
    if (p >= NPAIR) return;
    float t = 2.0f * FDIM + 1.0f;
    float disc = t * t - 8.0f * (float)p;
    int i = (int)((t - sqrtf(disc)) * 0.5f);
    if (i < 0) i = 0;
    if (i > FDIM - 1) i = FDIM - 1;
    while (i > 0 && (long long)i * (2 * FDIM - i + 1) / 2 > (long long)p) --i;
    while ((long long)(i + 1) * (2 * FDIM - i) / 2 <= (long long)p) ++i;
    long long off = (long long)i * (2 * FDIM - i + 1) / 2;
    int j = i + (int)((long long)p - off);
    pairs[p] = ((unsigned int)j << 16) | (unsigned int)i;
}

// ---------------------------------------------------------------------------
// Zero-pad dense features 1024x13 -> 1024x16 so all GEMM K are %4 == 0.
// ---------------------------------------------------------------------------
__global__ __launch_bounds__(256)
void pad_dense_kernel(const float* __restrict__ dense, float* __restrict__ out) {
    int idx = blockIdx.x * blockDim.x + threadIdx.x;
    if (idx >= BATCH * 16) return;
    int b = idx >> 4;
    int f = idx & 15;
    out[idx] = (f < NDENSE) ? dense[b * NDENSE + f] : 0.0f;
}

// ---------------------------------------------------------------------------
// Generic fp32 WMMA GEMM (compile-time K, K % 4 == 0, branch-free):
//   C[m,n] = sum_k A[m,k]*W[n,k] + bias[n]
// One wave per 16x16 tile. A-frag (ISA 16x4 f32): lane L -> m=L%16,
// K-pair offset 2*(L>=16); fragment halves are contiguous -> b64 loads.
// ---------------------------------------------------------------------------
template <int K>
__global__ __launch_bounds__(32)
void wmma_gemm_bias_kernel(const float* __restrict__ A,
                           const float* __restrict__ W,
                           const float* __restrict__ bias,
                           float* __restrict__ C,
                           int N) {
    const int mt   = blockIdx.x * 16;
    const int nt   = blockIdx.y * 16;
    const int lane = threadIdx.x;
    const int half = lane >> 4;
    const int l16  = lane & 15;
    const int kb   = 2 * half;

    const float* __restrict__ arow = A + (size_t)(mt + l16) * K + kb;
    const float* __restrict__ wrow = W + (size_t)(nt + l16) * K + kb;

    v8f acc = {};
#pragma unroll 4
    for (int k0 = 0; k0 < K; k0 += 4) {
        v2f a = *(const v2f*)(arow + k0);
        v2f b = *(const v2f*)(wrow + k0);
        acc = __builtin_amdgcn_wmma_f32_16x16x4_f32(
            false, a, false, b, (short)0, acc, false, false);
    }

    float bn = bias[nt + l16];
    float* __restrict__ ccol = C + nt + l16;
#pragma unroll
    for (int r = 0; r < 8; ++r) {
        int m = mt + r + 8 * half;
        ccol[(size_t)m * N] = acc[r] + bn;
    }
}

// ---------------------------------------------------------------------------
// Fused interaction + top-layer-0 GEMM.
// z[b,n] = bias[n] + sum_{k<16} feat[b,k]*W[n,k]
//                  + sum_p feat[b,i_p]*feat[b,j_p]*W[n,16+p]
// Block = 4 waves (4 N-tiles) sharing a 64-row feature stage in LDS (110 KB);
// each wave accumulates 4 M-tiles so each B fragment is reused 4x in regs.
// ---------------------------------------------------------------------------
__global__ __launch_bounds__(128)
void top0_fused_kernel(const float* __restrict__ feat,
                       const float* __restrict__ W,      // (512, KTOP0)
                       const float* __restrict__ bias,
                       const unsigned int* __restrict__ pairs,
                       float* __restrict__ C) {          // (1024, 512)
    __shared__ float sf[MT * 16 * FDIM];                  // 110,592 B

    const int mt0  = blockIdx.x * (MT * 16);
    const int wave = threadIdx.x >> 5;                    // 0..3
    const int lane = threadIdx.x & 31;
    const int nt   = (blockIdx.y * 4 + wave) * 16;
    const int half = lane >> 4;
    const int l16  = lane & 15;
    const int kb   = 2 * half;

    // Stage 64 contiguous feature rows into LDS as float4 (b128 traffic).
    {
        const float4* __restrict__ src =
            (const float4*)(feat + (size_t)mt0 * FDIM);
        float4* __restrict__ dst = (float4*)sf;
        for (int idx = threadIdx.x; idx < MT * 16 * FDIM / 4; idx += 128) {
            dst[idx] = src[idx];
        }
    }
    __syncthreads();

    const float* __restrict__ wrow = W + (size_t)(nt + l16) * KTOP0 + kb;
    const float* ar[MT];
#pragma unroll
    for (int t = 0; t < MT; ++t) ar[t] = sf + (t * 16 + l16) * FDIM;

    v8f acc[MT] = {};

    // Dense region: k in [0,16) (feat[0..15] == dense_out).
#pragma unroll
    for (int k0 = 0; k0 < 16; k0 += 4) {
        v2f b = *(const v2f*)(wrow + k0);
#pragma unroll
        for (int t = 0; t < MT; ++t) {
            v2f a;
            a.x = ar[t][k0 + kb];
            a.y = ar[t][k0 + kb + 1];
            acc[t] = __builtin_amdgcn_wmma_f32_16x16x4_f32(
                false, a, false, b, (short)0, acc[t], false, false);
        }
    }

    // Interaction region: k in [16, KTOP0); (KTOP0-16) % 4 == 0.
    // Unroll 4 => 16 WMMA per iteration with deep load clauses; prefetch the
    // weight stream ~2KB ahead once per 16-K chunk (global_prefetch_b8).
#pragma unroll 4
    for (int k0 = 16; k0 < KTOP0; k0 += 4) {
        if (((k0 - 16) & 12) == 0) {
            __builtin_prefetch(wrow + k0 + 512, 0, 1);
        }
        uint2 pp = *(const uint2*)(pairs + (k0 - 16 + kb));
        v2f  b  = *(const v2f*)(wrow + k0);
        int i0 = (int)(pp.x & 0xffffu), j0 = (int)(pp.x >> 16);
        int i1 = (int)(pp.y & 0xffffu), j1 = (int)(pp.y >> 16);
#pragma unroll
        for (int t = 0; t < MT; ++t) {
            v2f a;
            a.x = ar[t][i0] * ar[t][j0];
            a.y = ar[t][i1] * ar[t][j1];
            acc[t] = __builtin_amdgcn_wmma_f32_16x16x4_f32(
                false, a, false, b, (short)0, acc[t], false, false);
        }
    }

    float bn = bias[nt + l16];
    float* __restrict__ ccol = C + nt + l16;
#pragma unroll
    for (int t = 0; t < MT; ++t) {
#pragma unroll
        for (int r = 0; r < 8; ++r) {
            int m = mt0 + t * 16 + r + 8 * half;
            ccol[(size_t)m * 512] = acc[t][r] + bn;
        }
    }
}

// ---------------------------------------------------------------------------
// Column-wise BatchNorm (+ optional ReLU) in place, coalesced:
// each block handles 16 columns with a 16x16 thread tile so every 16-lane
// group reads 64B contiguous. N must be a multiple of 16.
// ---------------------------------------------------------------------------
__global__ __launch_bounds__(256)
void bn_relu_kernel(float* __restrict__ X,
                    const float* __restrict__ g,
                    const float* __restrict__ be,
                    int M, int N, int apply_relu) {
    __shared__ float ssum[16][17];
    __shared__ float ssq[16][17];
    const int tc = threadIdx.x & 15;          // column within group
    const int tr = threadIdx.x >> 4;          // row-thread
    const int n  = blockIdx.x * 16 + tc;

    float sum = 0.0f, sq = 0.0f;
    for (int m = tr; m < M; m += 16) {
        float v = X[(size_t)m * N + n];
        sum += v;
        sq  += v * v;
    }
    ssum[tr][tc] = sum;
    ssq[tr][tc]  = sq;
    __syncthreads();
    for (int s = 8; s > 0; s >>= 1) {
        if (tr < s) {
            ssum[tr][tc] += ssum[tr + s][tc];
            ssq[tr][tc]  += ssq[tr + s][tc];
        }
        __syncthreads();
    }
    float mean  = ssum[0][tc] / (float)M;
    float var   = ssq[0][tc] / (float)M - mean * mean;
    float scale = g[n] * rsqrtf(var + 1e-5f);
    float shift = be[n] - mean * scale;

    for (int m = tr; m < M; m += 16) {
        float v = X[(size_t)m * N + n] * scale + shift;
        X[(size_t)m * N + n] = apply_relu ? fmaxf(v, 0.0f) : v;
    }
}

// ---------------------------------------------------------------------------
// Embedding gather + feature assembly: feat[b] = [dense_out[b] | embs[b]].
// ---------------------------------------------------------------------------
__global__ __launch_bounds__(256)
void build_feat_kernel(const float* __restrict__ dense_out,
                       const long long* __restrict__ sparse,
                       const float* __restrict__ emb,
                       float* __restrict__ feat) {
    int idx = blockIdx.x * blockDim.x + threadIdx.x;
    if (idx >= BATCH * FDIM) return;
    int b = idx / FDIM;
    int f = idx % FDIM;
    float v;
    if (f < 16) {
        v = dense_out[b * 16 + f];
    } else {
        int t = (f - 16) / DEMB;
        int d = (f - 16) % DEMB;
        long long ix = sparse[(size_t)b * NTAB + t];
        v = emb[((long long)t * VOCAB + ix) * DEMB + d];
    }
    feat[idx] = v;
}

// ---------------------------------------------------------------------------
// Final layer: out[b] = dot(Y[b,:256], w_t2) + b_t2 (no BN).
// ---------------------------------------------------------------------------
__global__ __launch_bounds__(256)
void final_dot_kernel(const float* __restrict__ Y,
                      const float* __restrict__ w,
                      const float* __restrict__ b,
                      float* __restrict__ out) {
    __shared__ float s[256];
    const int row = blockIdx.x;
    const int t   = threadIdx.x;
    s[t] = Y[(size_t)row * 256 + t] * w[t];
    __syncthreads();
    for (int st = 128; st > 0; st >>= 1) {
        if (t < st) s[t] += s[t + st];
        __syncthreads();
    }
    if (t == 0) out[row] = s[0] + b[0];
}

// ---------------------------------------------------------------------------
// Host launch.
// ---------------------------------------------------------------------------
extern "C" void kernel_launch(void* const* d_in, const int* in_sizes, int n_in,
                              void* d_out, int out_size, void* d_ws, size_t ws_size,
                              hipStream_t stream) {
    (void)in_sizes; (void)n_in; (void)out_size; (void)ws_size;

    const float*      dense  = (const float*)d_in[0];
    const long long*  sparse = (const long long*)d_in[1];
    const float*      emb    = (const float*)d_in[2];
    const float *w_b0 = (const float*)d_in[3],  *b_b0 = (const float*)d_in[4];
    const float *g_b0 = (const float*)d_in[5],  *be_b0 = (const float*)d_in[6];
    const float *w_b1 = (const float*)d_in[7],  *b_b1 = (const float*)d_in[8];
    const float *g_b1 = (const float*)d_in[9],  *be_b1 = (const float*)d_in[10];
    const float *w_b2 = (const float*)d_in[11], *b_b2 = (const float*)d_in[12];
    const float *g_b2 = (const float*)d_in[13], *be_b2 = (const float*)d_in[14];
    const float *w_t0 = (const float*)d_in[15], *b_t0 = (const float*)d_in[16];
    const float *g_t0 = (const float*)d_in[17], *be_t0 = (const float*)d_in[18];
    const float *w_t1 = (const float*)d_in[19], *b_t1 = (const float*)d_in[20];
    const float *g_t1 = (const float*)d_in[21], *be_t1 = (const float*)d_in[22];
    const float *w_t2 = (const float*)d_in[23], *b_t2 = (const float*)d_in[24];
    float* out = (float*)d_out;

    // Workspace layout (fp32 elements; ~8.6 MB total).
    float* ws        = (float*)d_ws;
    float* dense_pad = ws;                          // 1024*16
    float* act0      = dense_pad + BATCH * 16;      // 1024*512
    float* act1      = act0 + BATCH * 512;          // 1024*256
    float* dense_out = act1 + BATCH * 256;          // 1024*16
    float* feat      = dense_out + BATCH * 16;      // 1024*432
    float* zt0       = feat + BATCH * FDIM;         // 1024*512
    float* zt1       = zt0 + BATCH * 512;           // 1024*256
    unsigned int* pairs = (unsigned int*)(zt1 + BATCH * 256);  // 93528 u32

    // Pair index table + zero-padded dense features.
    init_pairs_kernel<<<(NPAIR + 255) / 256, 256, 0, stream>>>(pairs);
    pad_dense_kernel<<<(BATCH * 16 + 255) / 256, 256, 0, stream>>>(dense, dense_pad);

    // Bottom MLP.
    wmma_gemm_bias_kernel<16><<<dim3(BATCH / 16, 512 / 16), 32, 0, stream>>>(
        dense_pad, w_b0, b_b0, act0, 512);
    bn_relu_kernel<<<512 / 16, 256, 0, stream>>>(act0, g_b0, be_b0, BATCH, 512, 1);

    wmma_gemm_bias_kernel<512><<<dim3(BATCH / 16, 256 / 16), 32, 0, stream>>>(
        act0, w_b1, b_b1, act1, 256);
    bn_relu_kernel<<<256 / 16, 256, 0, stream>>>(act1, g_b1, be_b1, BATCH, 256, 1);

    wmma_gemm_bias_kernel<256><<<dim3(BATCH / 16, 16 / 16), 32, 0, stream>>>(
        act1, w_b2, b_b2, dense_out, 16);
    bn_relu_kernel<<<1, 256, 0, stream>>>(dense_out, g_b2, be_b2, BATCH, 16, 1);

    // Embedding gather + feature assembly.
    build_feat_kernel<<<(BATCH * FDIM + 255) / 256, 256, 0, stream>>>(
        dense_out, sparse, emb, feat);

    // Fused interaction + top layer 0 (the 98-GFLOP GEMM).
    top0_fused_kernel<<<dim3(BATCH / (MT * 16), 512 / 64), 128, 0, stream>>>(
        feat, w_t0, b_t0, pairs, zt0);
    bn_relu_kernel<<<512 / 16, 256, 0, stream>>>(zt0, g_t0, be_t0, BATCH, 512, 1);

    // Top layer 1.
    wmma_gemm_bias_kernel<512><<<dim3(BATCH / 16, 256 / 16), 32, 0, stream>>>(
        zt0, w_t1, b_t1, zt1, 256);
    bn_relu_kernel<<<256 / 16, 256, 0, stream>>>(zt1, g_t1, be_t1, BATCH, 256, 1);

    // Final linear.
    final_dot_kernel<<<BATCH, 256, 0, stream>>>(zt1, w_t2, b_t2, out);
}